// GAT_66907000537778
// MI455X (gfx1250) — compile-verified
//
#include <hip/hip_runtime.h>
#include <hip/hip_bf16.h>
#include <math.h>

// ---------------------------------------------------------------------------
// GAT (3-layer) for MI455X / gfx1250.
// GEMMs use V_WMMA_F32_16X16X4_F32 (fp32 WMMA, wave32, 16x16 C tile / wave).
// Edge softmax/aggregation: wave-per-edge float4 gather + global f32 atomics.
// ---------------------------------------------------------------------------

typedef __attribute__((ext_vector_type(2))) float v2f;
typedef __attribute__((ext_vector_type(8))) float v8f;

#define N_NODES   50000
#define N_EDGES_C 800000
#define IN_SIZE   256
#define HIDDEN    128   // HEADS*HID
#define HEADS     4
#define HID       32
#define OUT_SIZE  40
#define OUT_PAD   48    // padded to 3 WMMA N-tiles; padding cols never read
#define NEG_SLOPE 0.2f

// order-preserving float <-> uint encoding for atomic max over signed floats
__device__ __forceinline__ unsigned fenc(float f) {
    unsigned u = __float_as_uint(f);
    return (u & 0x80000000u) ? ~u : (u | 0x80000000u);
}
__device__ __forceinline__ float fdec(unsigned u) {
    unsigned v = (u & 0x80000000u) ? (u & 0x7FFFFFFFu) : ~u;
    return __uint_as_float(v);
}
#define ENC_NEG_INF 0x007FFFFFu   // fenc(-inf)

// ---------------------------------------------------------------------------
// WMMA fp32 GEMM:  C[M x ldc] = A[M x K] @ B[K x Nact]   (row-major)
// One wave computes one 16x16 C tile; 8 waves / block sweep N tiles.
// M must be a multiple of 16 (50000 = 3125*16). K multiple of 4.
// For Nact < ldc (layer 3), out-of-range columns are CLAMPED, not zeroed:
// the padded columns of C receive junk that downstream kernels never read.
// This keeps the inner loop branchless (no EXEC masking around WMMA).
// ---------------------------------------------------------------------------
__global__ void gemm_wmma_f32(const float* __restrict__ A,
                              const float* __restrict__ B,
                              float* __restrict__ C,
                              int M, int K, int Nact, int ldc, int nTiles) {
    const int wave = threadIdx.x >> 5;
    const int lane = threadIdx.x & 31;
    const int tileM = blockIdx.x;
    const int tileN = wave;                 // 8 waves -> up to 8 N tiles
    if (tileN >= nTiles) return;            // wave-uniform; EXEC all-1 for WMMA

    const int m0   = tileM << 4;
    const int n0   = tileN << 4;
    const int half = lane >> 4;             // selects K pair {0,1} vs {2,3}
    const int r16  = lane & 15;             // A row / B col within tile

    const float* arow = A + (size_t)(m0 + r16) * K + (half << 1);
    int ncol = n0 + r16;
    ncol = (ncol < Nact) ? ncol : (Nact - 1);   // clamp: safe, padding never read
    const float* b0p = B + (size_t)(half << 1) * Nact + ncol;        // row 2*half
    const float* b1p = b0p + Nact;                                   // row 2*half+1
    const size_t bstep = (size_t)4 * Nact;

    v8f acc = {};
    for (int k0 = 0; k0 < K; k0 += 4) {
        v2f a;
        a.x = arow[0];
        a.y = arow[1];
        v2f b;
        b.x = *b0p;
        b.y = *b1p;
        acc = __builtin_amdgcn_wmma_f32_16x16x4_f32(
            /*neg_a=*/false, a, /*neg_b=*/false, b,
            /*c_mod=*/(short)0, acc, /*reuse_a=*/false, /*reuse_b=*/false);
        arow += 4;
        b0p  += bstep;
        b1p  += bstep;
    }

    // C/D layout: VGPR r -> row m0 + r + 8*(lane>>4), col n0 + (lane&15)
    float* crow = C + (size_t)(m0 + (half << 3)) * ldc + (n0 + r16);
    #pragma unroll
    for (int r = 0; r < 8; ++r)
        crow[(size_t)r * ldc] = acc[r];
}

// ---------------------------------------------------------------------------
// el[n,h] = sum_f z[n, h*F+f] * al[h*F+f]   (and er with ar)
// ---------------------------------------------------------------------------
__global__ void attn_dots(const float* __restrict__ z,
                          const float* __restrict__ al,
                          const float* __restrict__ ar,
                          float* __restrict__ el, float* __restrict__ er,
                          int nNodes, int H, int F, int ldz) {
    int idx = blockIdx.x * blockDim.x + threadIdx.x;
    if (idx >= nNodes * H) return;
    int n = idx / H, h = idx - n * H;
    const float* zp  = z + (size_t)n * ldz + h * F;
    const float* alp = al + h * F;
    const float* arp = ar + h * F;
    float sl = 0.f, sr = 0.f;
    for (int f = 0; f < F; ++f) {
        float v = zp[f];
        sl = fmaf(v, alp[f], sl);
        sr = fmaf(v, arp[f], sr);
    }
    el[idx] = sl;
    er[idx] = sr;
}

__global__ void fill_u32(unsigned* __restrict__ p, unsigned v, int n) {
    int i = blockIdx.x * blockDim.x + threadIdx.x;
    if (i < n) p[i] = v;
}

// ---------------------------------------------------------------------------
// Pass 1: e = leaky_relu(el[src]+er[dst]); segment-max into mEnc via atomicMax
// ---------------------------------------------------------------------------
__global__ void edge_e_max(const int* __restrict__ src, const int* __restrict__ dst,
                           const float* __restrict__ el, const float* __restrict__ er,
                           float* __restrict__ ebuf, unsigned* __restrict__ mEnc,
                           int E, int H) {
    int idx = blockIdx.x * blockDim.x + threadIdx.x;
    if (idx >= E * H) return;
    int e = idx / H, h = idx - e * H;
    int s = src[e], d = dst[e];
    float v = el[s * H + h] + er[d * H + h];
    v = (v > 0.f) ? v : NEG_SLOPE * v;
    ebuf[idx] = v;
    atomicMax(&mEnc[d * H + h], fenc(v));
}

// ---------------------------------------------------------------------------
// Pass 2: ee = exp(e - m[dst]); segment-sum into ssum via atomicAdd
// ---------------------------------------------------------------------------
__global__ void edge_exp_sum(const int* __restrict__ dst,
                             float* __restrict__ ebuf,
                             const unsigned* __restrict__ mEnc,
                             float* __restrict__ ssum, int E, int H) {
    int idx = blockIdx.x * blockDim.x + threadIdx.x;
    if (idx >= E * H) return;
    int e = idx / H, h = idx - e * H;
    int d = dst[e];
    float ee = expf(ebuf[idx] - fdec(mEnc[d * H + h]));
    ebuf[idx] = ee;
    atomicAdd(&ssum[d * H + h], ee);
}

// ---------------------------------------------------------------------------
// Aggregation, H=4/F=32 (128 floats): one wave32 per edge, float4 per lane.
// alpha computed in-flight: ee / ssum[dst].
// ---------------------------------------------------------------------------
__global__ void edge_aggregate_128(const int* __restrict__ src, const int* __restrict__ dst,
                                   const float* __restrict__ z,
                                   const float* __restrict__ ee, const float* __restrict__ ssum,
                                   float* __restrict__ out, int E) {
    int gw   = (blockIdx.x * blockDim.x + threadIdx.x) >> 5;
    int lane = threadIdx.x & 31;
    if (gw >= E) return;
    int s = src[gw], d = dst[gw];
    int i0 = lane << 2;           // 4 floats per lane, all in same head (i0>>5)
    int h  = i0 >> 5;
    float alpha = ee[gw * HEADS + h] / ssum[d * HEADS + h];
    const float4 zv = *reinterpret_cast<const float4*>(z + (size_t)s * HIDDEN + i0);
    float* op = out + (size_t)d * HIDDEN + i0;
    atomicAdd(op + 0, zv.x * alpha);
    atomicAdd(op + 1, zv.y * alpha);
    atomicAdd(op + 2, zv.z * alpha);
    atomicAdd(op + 3, zv.w * alpha);
}

// Aggregation, H=1 (output layer, F=40, z stride ldz=48)
__global__ void edge_aggregate_h1(const int* __restrict__ src, const int* __restrict__ dst,
                                  const float* __restrict__ z,
                                  const float* __restrict__ ee, const float* __restrict__ ssum,
                                  float* __restrict__ out, int E, int F, int ldz) {
    int idx = blockIdx.x * blockDim.x + threadIdx.x;
    if (idx >= E * F) return;
    int e = idx / F, f = idx - e * F;
    int s = src[e], d = dst[e];
    float alpha = ee[e] / ssum[d];
    atomicAdd(&out[(size_t)d * F + f], z[(size_t)s * ldz + f] * alpha);
}

// out = relu(agg + bias[c])
__global__ void bias_relu(const float* __restrict__ agg, const float* __restrict__ bias,
                          float* __restrict__ out, int total, int C) {
    int idx = blockIdx.x * blockDim.x + threadIdx.x;
    if (idx >= total) return;
    float v = agg[idx] + bias[idx % C];
    out[idx] = v > 0.f ? v : 0.f;
}

// in-place log_softmax over 40 classes, + bias; one wave per node
__global__ void log_softmax40(float* __restrict__ x, const float* __restrict__ bias,
                              int nNodes) {
    int node = (blockIdx.x * blockDim.x + threadIdx.x) >> 5;
    int lane = threadIdx.x & 31;
    if (node >= nNodes) return;
    float* row = x + (size_t)node * OUT_SIZE;
    const float NI = -INFINITY;
    float v0 = (lane < OUT_SIZE)      ? row[lane]      + bias[lane]      : NI;
    float v1 = (lane + 32 < OUT_SIZE) ? row[lane + 32] + bias[lane + 32] : NI;
    float mx = fmaxf(v0, v1);
    #pragma unroll
    for (int off = 16; off >= 1; off >>= 1)
        mx = fmaxf(mx, __shfl_xor(mx, off, 32));
    float se = ((lane < OUT_SIZE) ? expf(v0 - mx) : 0.f) +
               ((lane + 32 < OUT_SIZE) ? expf(v1 - mx) : 0.f);
    #pragma unroll
    for (int off = 16; off >= 1; off >>= 1)
        se += __shfl_xor(se, off, 32);
    float ls = logf(se);
    if (lane < OUT_SIZE)      row[lane]      = v0 - mx - ls;
    if (lane + 32 < OUT_SIZE) row[lane + 32] = v1 - mx - ls;
}

// ---------------------------------------------------------------------------
static inline int cdiv(long a, int b) { return (int)((a + b - 1) / b); }

extern "C" void kernel_launch(void* const* d_in, const int* in_sizes, int n_in,
                              void* d_out, int out_size, void* d_ws, size_t ws_size,
                              hipStream_t stream) {
    const float* X   = (const float*)d_in[0];
    const int*   src = (const int*)  d_in[1];
    const int*   dst = (const int*)  d_in[2];
    const float* W1  = (const float*)d_in[3];
    const float* al1 = (const float*)d_in[4];
    const float* ar1 = (const float*)d_in[5];
    const float* b1  = (const float*)d_in[6];
    const float* W2  = (const float*)d_in[7];
    const float* al2 = (const float*)d_in[8];
    const float* ar2 = (const float*)d_in[9];
    const float* b2  = (const float*)d_in[10];
    const float* W3  = (const float*)d_in[11];
    const float* al3 = (const float*)d_in[12];
    const float* ar3 = (const float*)d_in[13];
    const float* b3  = (const float*)d_in[14];

    const int N = in_sizes[0] / IN_SIZE;   // 50000
    const int E = in_sizes[1];             // 800000

    // workspace layout (floats)
    float*    Z   = (float*)d_ws;                         // [N,128] (or [N,48] L3)
    float*    Hh  = Z   + (size_t)N * HIDDEN;             // [N,128]
    float*    Agg = Hh  + (size_t)N * HIDDEN;             // [N,128]
    float*    el  = Agg + (size_t)N * HIDDEN;             // [N*4]
    float*    er  = el  + (size_t)N * HEADS;              // [N*4]
    unsigned* mEnc= (unsigned*)(er + (size_t)N * HEADS);  // [N*4]
    float*    ssum= (float*)(mEnc + (size_t)N * HEADS);   // [N*4]
    float*    eb  = ssum + (size_t)N * HEADS;             // [E*4]

    const int B256 = 256;
    const int mTiles = N / 16;             // 3125

    // ---------------- Layer 1 : in=256 -> 4 heads x 32 ----------------
    gemm_wmma_f32<<<mTiles, B256, 0, stream>>>(X, W1, Z, N, IN_SIZE, HIDDEN, HIDDEN, 8);
    attn_dots<<<cdiv((long)N * HEADS, B256), B256, 0, stream>>>(Z, al1, ar1, el, er, N, HEADS, HID, HIDDEN);
    fill_u32<<<cdiv((long)N * HEADS, B256), B256, 0, stream>>>(mEnc, ENC_NEG_INF, N * HEADS);
    hipMemsetAsync(ssum, 0, (size_t)N * HEADS * sizeof(float), stream);
    hipMemsetAsync(Agg,  0, (size_t)N * HIDDEN * sizeof(float), stream);
    edge_e_max  <<<cdiv((long)E * HEADS, B256), B256, 0, stream>>>(src, dst, el, er, eb, mEnc, E, HEADS);
    edge_exp_sum<<<cdiv((long)E * HEADS, B256), B256, 0, stream>>>(dst, eb, mEnc, ssum, E, HEADS);
    edge_aggregate_128<<<cdiv((long)E * 32, B256), B256, 0, stream>>>(src, dst, Z, eb, ssum, Agg, E);
    bias_relu<<<cdiv((long)N * HIDDEN, B256), B256, 0, stream>>>(Agg, b1, Hh, N * HIDDEN, HIDDEN);

    // ---------------- Layer 2 : 128 -> 4 heads x 32 ----------------
    gemm_wmma_f32<<<mTiles, B256, 0, stream>>>(Hh, W2, Z, N, HIDDEN, HIDDEN, HIDDEN, 8);
    attn_dots<<<cdiv((long)N * HEADS, B256), B256, 0, stream>>>(Z, al2, ar2, el, er, N, HEADS, HID, HIDDEN);
    fill_u32<<<cdiv((long)N * HEADS, B256), B256, 0, stream>>>(mEnc, ENC_NEG_INF, N * HEADS);
    hipMemsetAsync(ssum, 0, (size_t)N * HEADS * sizeof(float), stream);
    hipMemsetAsync(Agg,  0, (size_t)N * HIDDEN * sizeof(float), stream);
    edge_e_max  <<<cdiv((long)E * HEADS, B256), B256, 0, stream>>>(src, dst, el, er, eb, mEnc, E, HEADS);
    edge_exp_sum<<<cdiv((long)E * HEADS, B256), B256, 0, stream>>>(dst, eb, mEnc, ssum, E, HEADS);
    edge_aggregate_128<<<cdiv((long)E * 32, B256), B256, 0, stream>>>(src, dst, Z, eb, ssum, Agg, E);
    bias_relu<<<cdiv((long)N * HIDDEN, B256), B256, 0, stream>>>(Agg, b2, Hh, N * HIDDEN, HIDDEN);

    // ---------------- Layer 3 : 128 -> 1 head x 40, log_softmax ----------------
    float* out = (float*)d_out;
    gemm_wmma_f32<<<mTiles, B256, 0, stream>>>(Hh, W3, Z, N, HIDDEN, OUT_SIZE, OUT_PAD, 3);
    attn_dots<<<cdiv((long)N, B256), B256, 0, stream>>>(Z, al3, ar3, el, er, N, 1, OUT_SIZE, OUT_PAD);
    fill_u32<<<cdiv((long)N, B256), B256, 0, stream>>>(mEnc, ENC_NEG_INF, N);
    hipMemsetAsync(ssum, 0, (size_t)N * sizeof(float), stream);
    hipMemsetAsync(out,  0, (size_t)N * OUT_SIZE * sizeof(float), stream);
    edge_e_max  <<<cdiv((long)E, B256), B256, 0, stream>>>(src, dst, el, er, eb, mEnc, E, 1);
    edge_exp_sum<<<cdiv((long)E, B256), B256, 0, stream>>>(dst, eb, mEnc, ssum, E, 1);
    edge_aggregate_h1<<<cdiv((long)E * OUT_SIZE, B256), B256, 0, stream>>>(src, dst, Z, eb, ssum, out, E, OUT_SIZE, OUT_PAD);
    log_softmax40<<<cdiv((long)N * 32, B256), B256, 0, stream>>>(out, b3, N);
}